// SAE3DLSTM_70300024701092
// MI455X (gfx1250) — compile-verified
//
#include <hip/hip_runtime.h>

typedef float v2f __attribute__((ext_vector_type(2)));
typedef float v8f __attribute__((ext_vector_type(8)));

#define BN 2
#define CC 48
#define TT 4
#define HH 32
#define WW 32
#define THW 4096                 // TT*HH*WW
#define SZ (BN*CC*THW)           // 393216 floats per state tensor
#define NSTEP 4
#define LL 2
#define NHIST 3                  // c_hist depth = NSTEP-1
#define TO 2
#define HO 30
#define WO 30
#define CONVN (BN*TO*HO*WO)      // 3600 conv output positions
#define KC (CC*27)               // 1296: K per source tensor (divisible by 4)
#define KTOT (2*KC)              // 2592
#define KB 32                    // K-chunk staged in LDS per iteration
#define NCHUNK (KTOT/KB)         // 81
#define RCH 2
#define DIMQ 2048                // RC*H*W
#define NSEQ 6                   // agg, x, h, 3 hist
#define NROW (BN*NSEQ)           // 12

// ---------------------------------------------------------------- utilities
__global__ __launch_bounds__(256) void sae_zero(float* __restrict__ p, int n) {
    int i = blockIdx.x * 256 + threadIdx.x;
    if (i < n) p[i] = 0.f;
}

// ------------------------------------------------- implicit-GEMM conv (WMMA)
// Block = 96 threads (3 waves). N-panel = 48 positions (wave w -> n-tile w),
// full M = 48 (3 m-tiles per wave). The im2col B panel is staged into LDS in
// K-chunks of 32 with async global->LDS loads (ASYNCcnt), double-buffered so
// the gather for chunk i+1 is in flight while chunk i runs on the WMMA pipe.
__device__ __forceinline__ void sae_conv_epi(
    v8f acc, int mt, int mb, int ncol, const float* __restrict__ bias,
    float* __restrict__ out, int accum)
{
#pragma unroll
    for (int r = 0; r < 8; ++r) {
        int mm = mt * 16 + mb + r;
        float v = acc[r];
        if (bias) v += bias[mm];
        int o = mm * CONVN + ncol;
        if (accum) v += out[o];
        out[o] = v;
    }
}

__device__ __forceinline__ void sae_stage_async(
    int k0, unsigned lbase, const float* __restrict__ ssb0,
    const float* __restrict__ ssb1, int skc0, int snl)
{
#pragma unroll 1
    for (int j = 0; j < KB / 2; ++j) {
        int kc = skc0 + 2 * j;
        int kk = k0 + kc;
        const float* sp; int t;
        if (kk < KC) { sp = ssb0; t = kk; } else { sp = ssb1; t = kk - KC; }
        int c  = t / 27;  int r  = t - c  * 27;
        int kd = r / 9;   int r2 = r - kd * 9;
        int kh = r2 / 3;  int kw = r2 - kh * 3;
        const float* ga = sp + c * THW + kd * (HH * WW) + kh * WW + kw;
        unsigned la = lbase + (unsigned)((kc * 48 + snl) * 4);
        asm volatile("global_load_async_to_lds_b32 %0, %1, off"
                     :: "v"(la), "v"(ga) : "memory");
    }
}

__global__ __launch_bounds__(96) void sae_conv_wmma(
    const float* __restrict__ src0, const float* __restrict__ src1,
    const float* __restrict__ w0,   const float* __restrict__ w1,
    const float* __restrict__ bias, float* __restrict__ out, int accum)
{
    __shared__ float Bp[2 * KB * 48];             // 2 x 6 KB im2col panels
    const int tid  = threadIdx.x;
    const int lane = tid & 31;
    const int wid  = tid >> 5;                    // wave id 0..2 -> n-tile
    const int l16  = lane & 15;
    const int koff = (lane >> 4) << 1;            // lanes 16..31: K+2, K+3

    // ---- staging role: fixed n-column (tid%48), k rows tid/48 + 2j
    const int snl  = tid % 48;
    const int skc0 = tid / 48;                    // 0 or 1
    const int sn = blockIdx.x * 48 + snl;
    int swo = sn % WO; int stq = sn / WO;
    int sho = stq % HO; stq /= HO;
    int sto = stq % TO; int sb = stq / TO;
    const size_t sbase = (size_t)sb * CC * THW + sto * (HH * WW) + sho * WW + swo;
    const float* ssb0 = src0 + sbase;
    const float* ssb1 = src1 + sbase;
    const unsigned lds_base = (unsigned)(uintptr_t)(&Bp[0]);

    // ---- compute role
    const int ncol = blockIdx.x * 48 + wid * 16 + l16;
    const int nl   = wid * 16 + l16;              // column within LDS panel

    v8f acc0 = {}, acc1 = {}, acc2 = {};

    // prologue: stage chunk 0 into panel 0
    sae_stage_async(0, lds_base, ssb0, ssb1, skc0, snl);

    for (int i = 0; i < NCHUNK; ++i) {
        // complete this wave's in-flight panel, then publish across waves.
        // The same barrier guarantees all waves finished reading the panel
        // that the next stage call will overwrite.
        asm volatile("s_wait_asynccnt %0" :: "n"(0) : "memory");
        __syncthreads();

        const float* Bcur = &Bp[(i & 1) * (KB * 48)];
        if (i + 1 < NCHUNK)
            sae_stage_async((i + 1) * KB,
                            lds_base + (unsigned)(((i + 1) & 1) * (KB * 48 * 4)),
                            ssb0, ssb1, skc0, snl);

        const int k0 = i * KB;
#pragma unroll
        for (int kc = 0; kc < KB; kc += 4) {
            int kcl = kc + koff;
            v2f bv = { Bcur[kcl * 48 + nl], Bcur[(kcl + 1) * 48 + nl] };
            int kk = k0 + kc + koff;              // 4-aligned group: one source
            const float* wA; int t;
            if (kk < KC) { wA = w0; t = kk; } else { wA = w1; t = kk - KC; }
            v2f a0 = *(const v2f*)(wA + (size_t)(0 * 16 + l16) * KC + t);
            v2f a1 = *(const v2f*)(wA + (size_t)(1 * 16 + l16) * KC + t);
            v2f a2 = *(const v2f*)(wA + (size_t)(2 * 16 + l16) * KC + t);
            acc0 = __builtin_amdgcn_wmma_f32_16x16x4_f32(false, a0, false, bv,
                                                         (short)0, acc0, false, false);
            acc1 = __builtin_amdgcn_wmma_f32_16x16x4_f32(false, a1, false, bv,
                                                         (short)0, acc1, false, false);
            acc2 = __builtin_amdgcn_wmma_f32_16x16x4_f32(false, a2, false, bv,
                                                         (short)0, acc2, false, false);
        }
    }

    const int mb = (lane >= 16) ? 8 : 0;
    sae_conv_epi(acc0, 0, mb, ncol, bias, out, accum);
    sae_conv_epi(acc1, 1, mb, ncol, bias, out, accum);
    sae_conv_epi(acc2, 2, mb, ncol, bias, out, accum);
}

// ----------------------------- nearest-resize + _lr + activation (per b,c)
__global__ __launch_bounds__(256) void sae_gate(
    const float* __restrict__ zall, float* __restrict__ outall, unsigned tanh_mask)
{
    const int g = blockIdx.y;
    const float* z = zall + (size_t)g * CC * CONVN;
    float* out = outall + (size_t)g * SZ;
    const int b = blockIdx.x / CC;
    const int c = blockIdx.x % CC;
    const int tid = threadIdx.x;
    __shared__ float s1[256], s2[256];

    float a = 0.f, q = 0.f;
    for (int i = tid; i < THW; i += 256) {
        int t = i >> 10; int hw = i & 1023;
        int hr = hw >> 5; int wc = hw & 31;
        int ti = t >> 1;
        int hi = (hr * HO) >> 5;
        int wi = (wc * WO) >> 5;
        float v = z[c * CONVN + ((b * TO + ti) * HO + hi) * WO + wi];
        a += v; q += v * v;
    }
    s1[tid] = a; s2[tid] = q; __syncthreads();
    for (int st = 128; st > 0; st >>= 1) {
        if (tid < st) { s1[tid] += s1[tid + st]; s2[tid] += s2[tid + st]; }
        __syncthreads();
    }
    float mu  = s1[0] * (1.f / 4096.f);
    float var = s2[0] * (1.f / 4096.f) - mu * mu;
    float rs  = rsqrtf(var + 1e-5f);
    int is_tanh = (tanh_mask >> g) & 1u;
    for (int i = tid; i < THW; i += 256) {
        int t = i >> 10; int hw = i & 1023;
        int hr = hw >> 5; int wc = hw & 31;
        int ti = t >> 1;
        int hi = (hr * HO) >> 5;
        int wi = (wc * WO) >> 5;
        float v = z[c * CONVN + ((b * TO + ti) * HO + hi) * WO + wi];
        float u = (v - mu) * rs;
        out[((size_t)b * CC + c) * THW + i] = is_tanh ? tanhf(u)
                                                      : 1.f / (1.f + expf(-u));
    }
}

// ---------------------------------- attention: channel/time reduction -> seq
__global__ __launch_bounds__(256) void sae_red(
    const float* __restrict__ agg, const float* __restrict__ xt, const float* __restrict__ h,
    const float* __restrict__ ch0, const float* __restrict__ ch1, const float* __restrict__ ch2,
    const float* __restrict__ wr, const float* __restrict__ br, float* __restrict__ seq)
{
    int id = blockIdx.x;
    int j = id / (BN * RCH); int rem = id % (BN * RCH);
    int b = rem / RCH; int r = rem % RCH;
    const float* src;
    switch (j) {
        case 0:  src = agg; break;                              // broadcast over b
        case 1:  src = xt  + (size_t)b * CC * THW; break;
        case 2:  src = h   + (size_t)b * CC * THW; break;
        case 3:  src = ch0 + (size_t)b * CC * THW; break;
        case 4:  src = ch1 + (size_t)b * CC * THW; break;
        default: src = ch2 + (size_t)b * CC * THW; break;
    }
    __shared__ float swr[CC * TT];
    for (int i = threadIdx.x; i < CC * TT; i += 256) swr[i] = wr[r * CC * TT + i];
    __syncthreads();
    float bias = br[r];
    for (int hw = threadIdx.x; hw < HH * WW; hw += 256) {
        float acc = bias;
        for (int c = 0; c < CC; ++c) {
            const float* sc = src + c * THW + hw;
#pragma unroll
            for (int t = 0; t < TT; ++t) acc += sc[t * (HH * WW)] * swr[c * TT + t];
        }
        seq[((size_t)(b * NSEQ + j)) * DIMQ + r * (HH * WW) + hw] = acc;
    }
}

// ----------------------- qk projection: out[4096,16] = w_qk[4096,2048] @ seq^T
__global__ __launch_bounds__(32) void sae_qk_wmma(
    const float* __restrict__ wqk, const float* __restrict__ seq, float* __restrict__ out)
{
    const int lane = threadIdx.x;
    const int m0 = blockIdx.x * 16;
    const int l16  = lane & 15;
    const int koff = (lane >> 4) << 1;
    const int mrow = m0 + l16;
    const bool valid = l16 < NROW;
    v8f acc = {};
#pragma unroll 1
    for (int k = 0; k < DIMQ; k += 4) {
        int kk = k + koff;
        v2f av = *(const v2f*)(wqk + (size_t)mrow * DIMQ + kk);
        float b0 = valid ? seq[(size_t)l16 * DIMQ + kk]     : 0.f;
        float b1 = valid ? seq[(size_t)l16 * DIMQ + kk + 1] : 0.f;
        v2f bv = {b0, b1};
        acc = __builtin_amdgcn_wmma_f32_16x16x4_f32(false, av, false, bv,
                                                    (short)0, acc, false, false);
    }
    const int mb = (lane >= 16) ? 8 : 0;
#pragma unroll
    for (int r = 0; r < 8; ++r)
        out[(size_t)(m0 + mb + r) * 16 + l16] = acc[r];
}

// ------------------------------------------- logits (q0 . k_j) and softmax
__global__ __launch_bounds__(256) void sae_attn_logits(
    const float* __restrict__ qk, float* __restrict__ attn)
{
    int b = blockIdx.x;
    int tid = threadIdx.x;
    float p[NSEQ];
#pragma unroll
    for (int j = 0; j < NSEQ; ++j) p[j] = 0.f;
    for (int d = tid; d < DIMQ; d += 256) {
        float qv = qk[(size_t)d * 16 + b * NSEQ];          // q row = seq row (b, j=0)
#pragma unroll
        for (int j = 0; j < NSEQ; ++j)
            p[j] += qv * qk[(size_t)(DIMQ + d) * 16 + b * NSEQ + j];
    }
    __shared__ float sh[NSEQ][256];
#pragma unroll
    for (int j = 0; j < NSEQ; ++j) sh[j][tid] = p[j];
    __syncthreads();
    for (int st = 128; st > 0; st >>= 1) {
        if (tid < st)
#pragma unroll
            for (int j = 0; j < NSEQ; ++j) sh[j][tid] += sh[j][tid + st];
        __syncthreads();
    }
    if (tid == 0) {
        float scale = rsqrtf((float)DIMQ);
        float l[NSEQ], mx = -1e30f;
#pragma unroll
        for (int j = 0; j < NSEQ; ++j) { l[j] = sh[j][0] * scale; mx = fmaxf(mx, l[j]); }
        float e[NSEQ], s = 0.f;
#pragma unroll
        for (int j = 0; j < NSEQ; ++j) { e[j] = expf(l[j] - mx); s += e[j]; }
        float inv = 1.f / s;
#pragma unroll
        for (int j = 0; j < NSEQ; ++j) attn[b * NSEQ + j] = e[j] * inv;
    }
}

__global__ __launch_bounds__(256) void sae_attn_combine(
    const float* __restrict__ attn, const float* __restrict__ agg, const float* __restrict__ xt,
    const float* __restrict__ h, const float* __restrict__ ch0, const float* __restrict__ ch1,
    const float* __restrict__ ch2, float* __restrict__ s)
{
    int idx = blockIdx.x * 256 + threadIdx.x;
    int b = idx / (CC * THW);
    int rest = idx - b * (CC * THW);
    const float* a = attn + b * NSEQ;
    s[idx] = a[0] * agg[rest] + a[1] * xt[idx] + a[2] * h[idx]
           + a[3] * ch0[idx]  + a[4] * ch1[idx] + a[5] * ch2[idx];
}

// ---------------------------------- _ln4 stats: deterministic 2-pass reduce
__global__ __launch_bounds__(256) void sae_ln4_partial(
    const float* __restrict__ chl, const float* __restrict__ s, float* __restrict__ part)
{
    int tid = threadIdx.x;
    int idx = blockIdx.x * 256 + tid;
    float v = chl[idx] + s[idx];
    __shared__ float s1[256], s2[256];
    s1[tid] = v; s2[tid] = v * v; __syncthreads();
    for (int st = 128; st > 0; st >>= 1) {
        if (tid < st) { s1[tid] += s1[tid + st]; s2[tid] += s2[tid + st]; }
        __syncthreads();
    }
    if (tid == 0) { part[blockIdx.x * 2] = s1[0]; part[blockIdx.x * 2 + 1] = s2[0]; }
}

__global__ __launch_bounds__(256) void sae_ln4_final(
    const float* __restrict__ part, float* __restrict__ st)
{
    const int PB = (CC * THW) / 256;   // 768 partial blocks per batch
    int b = blockIdx.x, tid = threadIdx.x;
    float a = 0.f, q = 0.f;
    for (int i = tid; i < PB; i += 256) {
        a += part[(b * PB + i) * 2];
        q += part[(b * PB + i) * 2 + 1];
    }
    __shared__ float s1[256], s2[256];
    s1[tid] = a; s2[tid] = q; __syncthreads();
    for (int stp = 128; stp > 0; stp >>= 1) {
        if (tid < stp) { s1[tid] += s1[tid + stp]; s2[tid] += s2[tid + stp]; }
        __syncthreads();
    }
    if (tid == 0) {
        float inv = 1.f / (float)(CC * THW);
        float mu = s1[0] * inv;
        float var = s2[0] * inv - mu * mu;
        st[b * 2] = mu;
        st[b * 2 + 1] = rsqrtf(var + 1e-5f);
    }
}

// ------------------------------------------------ c = i*g + ln4 ; m = ip*gp+fp*m
__global__ __launch_bounds__(256) void sae_c_m2(
    const float* __restrict__ chl, const float* __restrict__ s, const float* __restrict__ st,
    const float* __restrict__ it, const float* __restrict__ gt, const float* __restrict__ ip,
    const float* __restrict__ gp, const float* __restrict__ fp,
    const float* __restrict__ lng, const float* __restrict__ lnb,
    float* __restrict__ cnew, float* __restrict__ m)
{
    int idx = blockIdx.x * 256 + threadIdx.x;
    int b = idx / (CC * THW);
    int rest = idx - b * (CC * THW);
    float z = chl[idx] + s[idx];
    float u = (z - st[b * 2]) * st[b * 2 + 1];
    cnew[idx] = it[idx] * gt[idx] + u * lng[rest] + lnb[rest];
    m[idx] = ip[idx] * gp[idx] + fp[idx] * m[idx];
}

// --------------------------------------- h = o * tanh(w111 @ [c;m] + b111)
__global__ __launch_bounds__(256) void sae_h2(
    const float* __restrict__ c, const float* __restrict__ m, const float* __restrict__ og,
    const float* __restrict__ w111, const float* __restrict__ b111, float* __restrict__ h)
{
    int idx = blockIdx.x * 256 + threadIdx.x;
    int pos = idx % THW;
    int t = idx / THW;
    int oc = t % CC;
    int b = t / CC;
    const float* cb = c + (size_t)b * CC * THW + pos;
    const float* mb = m + (size_t)b * CC * THW + pos;
    const float* wr = w111 + oc * 2 * CC;
    float acc = b111[oc];
#pragma unroll 4
    for (int j = 0; j < CC; ++j)
        acc += cb[j * THW] * wr[j] + mb[j * THW] * wr[CC + j];
    h[idx] = og[idx] * tanhf(acc);
}

// ============================================================== host driver
extern "C" void kernel_launch(void* const* d_in, const int* in_sizes, int n_in,
                              void* d_out, int out_size, void* d_ws, size_t ws_size,
                              hipStream_t stream)
{
    const float* x    = (const float*)d_in[0];
    const float* agg  = (const float*)d_in[1];
    const float* wx   = (const float*)d_in[2];
    const float* bx   = (const float*)d_in[3];
    const float* wh   = (const float*)d_in[4];
    const float* wred = (const float*)d_in[5];
    const float* bred = (const float*)d_in[6];
    const float* wqk  = (const float*)d_in[7];
    const float* lng  = (const float*)d_in[8];
    const float* lnb  = (const float*)d_in[9];
    const float* w111 = (const float*)d_in[10];
    const float* b111 = (const float*)d_in[11];
    (void)in_sizes; (void)n_in; (void)out_size; (void)ws_size;

    float* W  = (float*)d_ws;
    float* CH = W;                                    // L*NHIST*SZ c_hist ring
    float* HB = CH  + (size_t)LL * NHIST * SZ;        // L*SZ hidden states
    float* MB = HB  + (size_t)LL * SZ;                // SZ memory state
    float* ZB = MB  + (size_t)SZ;                     // 6*CC*CONVN conv outputs
    float* GB = ZB  + (size_t)6 * CC * CONVN;         // 6*SZ gates i,g,ip,gp,fp,o
    float* SB = GB  + (size_t)6 * SZ;                 // SZ attention output
    float* SEQ = SB + (size_t)SZ;                     // 12*2048
    float* QK  = SEQ + (size_t)NROW * DIMQ;           // 4096*16
    float* AT  = QK  + (size_t)4096 * 16;             // 16 attn weights
    float* ST  = AT  + 16;                            // 8 ln4 stats
    float* PT  = ST  + 8;                             // SZ/256*2 partials

    const int stateN = LL * NHIST * SZ + LL * SZ + SZ;
    sae_zero<<<(stateN + 255) / 256, 256, 0, stream>>>(CH, stateN);

    int ring[LL] = {0, 0};
    for (int step = 0; step < NSTEP; ++step) {
        const float* xt = x + (size_t)step * SZ;
        for (int l = 0; l < LL; ++l) {
            float* h = HB + (size_t)l * SZ;
            const float* wx_l = wx + (size_t)l * 6 * CC * KC;
            const float* wh_l = wh + (size_t)l * 8 * CC * KC;
            const float* bx_l = bx + (size_t)l * 6 * CC;
            float* ch[NHIST];
            for (int i = 0; i < NHIST; ++i)
                ch[i] = CH + ((size_t)l * NHIST + (size_t)((ring[l] + i) % NHIST)) * SZ;

            // pass 1: the six paired convolutions (x-conv + h/m-conv + bias)
            for (int g = 0; g < 6; ++g) {
                const float* s1 = (g >= 2 && g <= 4) ? MB : h;
                sae_conv_wmma<<<CONVN / 48, 96, 0, stream>>>(
                    xt, s1, wx_l + (size_t)g * CC * KC, wh_l + (size_t)g * CC * KC,
                    bx_l + g * CC, ZB + (size_t)g * CC * CONVN, 0);
            }
            // gates i_t, g_t, i_p, g_p, f_p  (tanh on gates 1 and 3)
            sae_gate<<<dim3(BN * CC, 5), 256, 0, stream>>>(ZB, GB, 0xAu);
            // attention
            sae_red<<<NSEQ * BN * RCH, 256, 0, stream>>>(
                agg, xt, h, ch[0], ch[1], ch[2],
                wred + (size_t)l * RCH * CC * TT, bred + (size_t)l * RCH, SEQ);
            sae_qk_wmma<<<4096 / 16, 32, 0, stream>>>(
                wqk + (size_t)l * 2 * DIMQ * DIMQ, SEQ, QK);
            sae_attn_logits<<<BN, 256, 0, stream>>>(QK, AT);
            sae_attn_combine<<<SZ / 256, 256, 0, stream>>>(
                AT, agg, xt, h, ch[0], ch[1], ch[2], SB);
            // _ln4 stats on c_hist[-1] + s
            sae_ln4_partial<<<SZ / 256, 256, 0, stream>>>(ch[NHIST - 1], SB, PT);
            sae_ln4_final<<<BN, 256, 0, stream>>>(PT, ST);
            // c (into oldest history slot) and m2 (in place)
            float* cnew = ch[0];
            sae_c_m2<<<SZ / 256, 256, 0, stream>>>(
                ch[NHIST - 1], SB, ST,
                GB, GB + (size_t)SZ, GB + (size_t)2 * SZ, GB + (size_t)3 * SZ,
                GB + (size_t)4 * SZ,
                lng + (size_t)l * CC * THW, lnb + (size_t)l * CC * THW, cnew, MB);
            // pass 2: accumulate cdh6(c)+cdh7(m2) into the o-gate conv buffer
            sae_conv_wmma<<<CONVN / 48, 96, 0, stream>>>(
                cnew, MB, wh_l + (size_t)6 * CC * KC, wh_l + (size_t)7 * CC * KC,
                nullptr, ZB + (size_t)5 * CC * CONVN, 1);
            sae_gate<<<dim3(BN * CC, 1), 256, 0, stream>>>(
                ZB + (size_t)5 * CC * CONVN, GB + (size_t)5 * SZ, 0u);
            // h2 = o * tanh(w111 @ [c;m2] + b111), written in place into h[l]
            sae_h2<<<SZ / 256, 256, 0, stream>>>(
                cnew, MB, GB + (size_t)5 * SZ,
                w111 + (size_t)l * CC * 2 * CC, b111 + (size_t)l * CC, h);

            ring[l] = (ring[l] + 1) % NHIST;
            xt = h;
        }
    }
    hipMemcpyAsync(d_out, HB + (size_t)(LL - 1) * SZ, (size_t)SZ * sizeof(float),
                   hipMemcpyDeviceToDevice, stream);
}